// Loss_36266703847560
// MI455X (gfx1250) — compile-verified
//
#include <hip/hip_runtime.h>
#include <hip/hip_bf16.h>
#include <math.h>

#define N_B   4
#define A_CNT 8732
#define C_CNT 81
#define NB3   ((A_CNT + 255) / 256)   // 35 blocks of 256 over anchors
#define NB2   ((A_CNT + 127) / 128)   // 69 blocks of 128 p-anchors (8 waves x 16)
#define GCH   1024                    // g-box LDS chunk
#define RCH   2048                    // con_neg LDS chunk
#define EPS_IOU 2.2204460492503131e-16f
#define FBIG  3.4e38f

typedef float v2f __attribute__((ext_vector_type(2)));
typedef float v8f __attribute__((ext_vector_type(8)));

__global__ void init_kernel(int* pos_num) {
  int t = threadIdx.x;
  if (t < N_B) pos_num[t] = 0;
}

// ---------------- Kernel 1: per-anchor elementwise prep ----------------
__global__ void prep_kernel(const float* __restrict__ ploc,
                            const float* __restrict__ plabel,
                            const float* __restrict__ gloc,
                            const int*   __restrict__ glabel,
                            const float* __restrict__ dboxes,
                            float* __restrict__ sl1,
                            float* __restrict__ con,
                            float* __restrict__ con_neg,
                            float* __restrict__ validf,
                            float* __restrict__ pL, float* __restrict__ pT,
                            float* __restrict__ pR, float* __restrict__ pB,
                            float* __restrict__ gL, float* __restrict__ gT,
                            float* __restrict__ gR, float* __restrict__ gB,
                            int* __restrict__ pos_num)
{
  const int n  = blockIdx.y;
  const int a  = blockIdx.x * blockDim.x + threadIdx.x;
  const bool act = (a < A_CNT);
  const int ai = act ? a : (A_CNT - 1);
  const int idx = n * A_CNT + ai;

  const float dx = dboxes[0*A_CNT + ai];
  const float dy = dboxes[1*A_CNT + ai];
  const float dw = dboxes[2*A_CNT + ai];
  const float dh = dboxes[3*A_CNT + ai];

  const float* pp = ploc + (size_t)n * 4 * A_CNT;
  const float* gg = gloc + (size_t)n * 4 * A_CNT;
  const float p0 = pp[0*A_CNT+ai], p1 = pp[1*A_CNT+ai];
  const float p2 = pp[2*A_CNT+ai], p3 = pp[3*A_CNT+ai];
  const float g0 = gg[0*A_CNT+ai], g1 = gg[1*A_CNT+ai];
  const float g2 = gg[2*A_CNT+ai], g3 = gg[3*A_CNT+ai];

  // vec_gd + smooth-L1 over 4 coords (IEEE division: count is tiny)
  const float t0 = 10.0f * (g0 - dx) / dw;
  const float t1 = 10.0f * (g1 - dy) / dh;
  const float t2 =  5.0f * logf(g2 / dw);
  const float t3 =  5.0f * logf(g3 / dh);
  auto sm = [](float d) { float ad = fabsf(d); return (ad < 1.0f) ? 0.5f*d*d : ad - 0.5f; };
  const float s = sm(p0 - t0) + sm(p1 - t1) + sm(p2 - t2) + sm(p3 - t3);

  // log-softmax over C (two-pass, stable) -> con at glabel
  const float* pc = plabel + (size_t)n * C_CNT * A_CNT + ai;
  float mx = -FBIG;
  #pragma unroll 1
  for (int c = 0; c < C_CNT; ++c) mx = fmaxf(mx, pc[(size_t)c * A_CNT]);
  float se = 0.0f;
  #pragma unroll 1
  for (int c = 0; c < C_CNT; ++c) se += expf(pc[(size_t)c * A_CNT] - mx);
  const int lab = glabel[idx];
  const float cn = (mx + logf(se)) - pc[(size_t)lab * A_CNT];
  const bool msk = (lab > 0);

  // decode ploc -> p_ltrb
  const float pcx = 0.1f * p0 * dw + dx;
  const float pcy = 0.1f * p1 * dh + dy;
  const float pw  = expf(0.2f * p2) * dw;
  const float ph  = expf(0.2f * p3) * dh;
  const float pl_ = pcx - 0.5f*pw, pt_ = pcy - 0.5f*ph;
  const float pr_ = pcx + 0.5f*pw, pb_ = pcy + 0.5f*ph;

  // decode gloc -> g_ltrb (reference decodes gloc identically)
  const float gcx = 0.1f * g0 * dw + dx;
  const float gcy = 0.1f * g1 * dh + dy;
  const float gw  = expf(0.2f * g2) * dw;
  const float gh  = expf(0.2f * g3) * dh;
  const float gl_ = gcx - 0.5f*gw, gt_ = gcy - 0.5f*gh;
  const float gr_ = gcx + 0.5f*gw, gb_ = gcy + 0.5f*gh;

  if (act) {
    sl1[idx]     = s;
    con[idx]     = cn;
    con_neg[idx] = msk ? 0.0f : cn;
    validf[idx]  = ((pl_ < pr_) && (pt_ < pb_)) ? 1.0f : 0.0f;
    pL[idx] = pl_; pT[idx] = pt_; pR[idx] = pr_; pB[idx] = pb_;
    gL[idx] = gl_; gT[idx] = gt_; gR[idx] = gr_; gB[idx] = gb_;
  }

  // positive count (integer atomics: deterministic)
  __shared__ int scnt;
  if (threadIdx.x == 0) scnt = 0;
  __syncthreads();
  if (act && msk) atomicAdd(&scnt, 1);
  __syncthreads();
  if (threadIdx.x == 0) atomicAdd(&pos_num[n], scnt);
}

// ---------------- Kernel 2: pairwise IoU sum with WMMA k-reduction ----------------
// Wave = 16 p-anchors; per step lane computes 2 IoU values laid out as the
// 16x4 f32 A-matrix; B = ones(4x16) so D accumulates row sums on the XDL.
// inter/ua uses v_rcp_f32 (TRANS, co-executes with VALU) instead of the
// ~10-instruction IEEE divide sequence: ~1 ulp, safe under tolerance since
// results are summed over 8732 terms in a different order than XLA anyway.
__global__ void iou_kernel(const float* __restrict__ pL, const float* __restrict__ pT,
                           const float* __restrict__ pR, const float* __restrict__ pB,
                           const float* __restrict__ gL, const float* __restrict__ gT,
                           const float* __restrict__ gR, const float* __restrict__ gB,
                           float* __restrict__ iou_s)
{
  __shared__ float sgl[GCH], sgt[GCH], sgr[GCH], sgb[GCH], sga[GCH];
  __shared__ float rowsum[8][16];

  const int n    = blockIdx.y;
  const int wave = threadIdx.x >> 5;
  const int lane = threadIdx.x & 31;
  const int m    = lane & 15;
  const int kb   = (lane >> 4) << 1;  // lanes 0-15 -> K{0,1}; lanes 16-31 -> K{2,3}
  const int pa   = blockIdx.x * 128 + wave * 16 + m;
  const bool pv  = (pa < A_CNT);
  const int pidx = n * A_CNT + (pv ? pa : 0);

  float pl = pL[pidx], pt = pT[pidx], pr = pR[pidx], pb = pB[pidx];
  float parea = (pr - pl) * (pb - pt);
  if (!pv) { pl = FBIG; pt = FBIG; pr = -FBIG; pb = -FBIG; parea = 0.0f; }

  v8f acc = {0.f, 0.f, 0.f, 0.f, 0.f, 0.f, 0.f, 0.f};
  v2f bones; bones.x = 1.0f; bones.y = 1.0f;

  for (int c0 = 0; c0 < A_CNT; c0 += GCH) {
    __syncthreads();
    #pragma unroll
    for (int t = threadIdx.x; t < GCH; t += 256) {
      int g = c0 + t;
      if (g < A_CNT) {
        int gi = n * A_CNT + g;
        float l = gL[gi], tt = gT[gi], r = gR[gi], b = gB[gi];
        sgl[t] = l; sgt[t] = tt; sgr[t] = r; sgb[t] = b;
        sga[t] = (r - l) * (b - tt);
      } else {
        sgl[t] = FBIG; sgt[t] = FBIG; sgr[t] = -FBIG; sgb[t] = -FBIG; sga[t] = 0.0f;
      }
    }
    __syncthreads();

    #pragma unroll 4
    for (int k = 0; k < GCH; k += 4) {
      v2f av;
      {
        int k0 = k + kb;
        float iw = fmaxf(fminf(pr, sgr[k0]) - fmaxf(pl, sgl[k0]), 0.0f);
        float ih = fmaxf(fminf(pb, sgb[k0]) - fmaxf(pt, sgt[k0]), 0.0f);
        float inter = iw * ih;
        float ua = fmaxf(parea + sga[k0] - inter, EPS_IOU);
        av.x = inter * __builtin_amdgcn_rcpf(ua);
      }
      {
        int k1 = k + kb + 1;
        float iw = fmaxf(fminf(pr, sgr[k1]) - fmaxf(pl, sgl[k1]), 0.0f);
        float ih = fmaxf(fminf(pb, sgb[k1]) - fmaxf(pt, sgt[k1]), 0.0f);
        float inter = iw * ih;
        float ua = fmaxf(parea + sga[k1] - inter, EPS_IOU);
        av.y = inter * __builtin_amdgcn_rcpf(ua);
      }
      // D = A(16x4) x ones(4x16) + C : row sums accumulate, every column equal
      acc = __builtin_amdgcn_wmma_f32_16x16x4_f32(
          false, av, false, bones, (short)0, acc, false, false);
    }
  }

  // Extract row sums: VGPR r holds M=r (lanes 0-15) and M=r+8 (lanes 16-31)
  #pragma unroll
  for (int r = 0; r < 8; ++r) {
    if (lane == r)      rowsum[wave][r]     = acc[r];
    if (lane == 16 + r) rowsum[wave][8 + r] = acc[r];
  }
  __syncthreads();
  if (lane < 16) {
    int pa2 = blockIdx.x * 128 + wave * 16 + lane;
    if (pa2 < A_CNT) iou_s[n * A_CNT + pa2] = rowsum[wave][lane];
  }
}

// ---------------- Kernel 3: exact stable rank + per-block partial losses ----------------
__global__ void rank_loss_kernel(const float* __restrict__ con,
                                 const float* __restrict__ con_neg,
                                 const float* __restrict__ sl1,
                                 const float* __restrict__ validf,
                                 const float* __restrict__ iou_s,
                                 const int*   __restrict__ glabel,
                                 const int*   __restrict__ pos_num,
                                 float* __restrict__ sl1part,
                                 float* __restrict__ clpart)
{
  __shared__ float scn[RCH];
  __shared__ float red1[256], red2[256];

  const int n   = blockIdx.y;
  const int tid = threadIdx.x;
  const int a   = blockIdx.x * 256 + tid;
  const bool act = (a < A_CNT);
  const int ai  = act ? a : 0;
  const int idx = n * A_CNT + ai;

  const float cna = con_neg[idx];
  int cnt = 0;

  for (int c0 = 0; c0 < A_CNT; c0 += RCH) {
    __syncthreads();
    #pragma unroll
    for (int t = tid; t < RCH; t += 256) {
      int b = c0 + t;
      scn[t] = (b < A_CNT) ? con_neg[n * A_CNT + b] : -FBIG;  // never counted (cna >= 0)
    }
    __syncthreads();
    #pragma unroll 4
    for (int t = 0; t < RCH; ++t) {
      float cb = scn[t];
      int b = c0 + t;
      cnt += (cb > cna) ? 1 : 0;
      cnt += ((cb == cna) && (b < a)) ? 1 : 0;   // stable-sort tie break
    }
  }

  const int pn = pos_num[n];
  const int neg_num = min(3 * pn, A_CNT);
  const int lab = glabel[idx];
  const bool msk = (lab > 0);
  const bool sel = (cnt < neg_num);

  float closs_c = 0.0f, sl1_c = 0.0f;
  if (act) {
    const float c = con[idx];
    closs_c = c * ((msk ? 1.0f : 0.0f) + (sel ? 1.0f : 0.0f));
    if (msk) {
      const float sv = sl1[idx];
      const float w = (validf[idx] != 0.0f) ? 0.01f * iou_s[idx] : 0.0f;
      sl1_c = sv / (sv + w) * sv;   // IEEE divide: negligible count
    }
  }

  // deterministic block tree-reduction
  red1[tid] = sl1_c; red2[tid] = closs_c;
  __syncthreads();
  for (int s = 128; s > 0; s >>= 1) {
    if (tid < s) { red1[tid] += red1[tid + s]; red2[tid] += red2[tid + s]; }
    __syncthreads();
  }
  if (tid == 0) {
    sl1part[n * NB3 + blockIdx.x] = red1[0];
    clpart[n * NB3 + blockIdx.x]  = red2[0];
  }
}

// ---------------- Kernel 4: final deterministic combine ----------------
__global__ void final_kernel(const int* __restrict__ pos_num,
                             const float* __restrict__ sl1part,
                             const float* __restrict__ clpart,
                             float* __restrict__ out)
{
  if (threadIdx.x == 0 && blockIdx.x == 0) {
    float acc = 0.0f;
    for (int n = 0; n < N_B; ++n) {
      float sp = 0.0f, cp = 0.0f;
      for (int b = 0; b < NB3; ++b) { sp += sl1part[n*NB3+b]; cp += clpart[n*NB3+b]; }
      float total = sp + cp;
      float nm = (pos_num[n] > 0) ? 1.0f : 0.0f;
      float pnv = fmaxf((float)pos_num[n], 1e-6f);
      acc += total * nm / pnv;
    }
    out[0] = acc * (1.0f / (float)N_B);
  }
}

extern "C" void kernel_launch(void* const* d_in, const int* in_sizes, int n_in,
                              void* d_out, int out_size, void* d_ws, size_t ws_size,
                              hipStream_t stream) {
  (void)in_sizes; (void)n_in; (void)out_size; (void)ws_size;
  const float* ploc   = (const float*)d_in[0];
  const float* plabel = (const float*)d_in[1];
  const float* gloc   = (const float*)d_in[2];
  const int*   glabel = (const int*)d_in[3];
  const float* dboxes = (const float*)d_in[4];
  float* out = (float*)d_out;

  const size_t NA = (size_t)N_B * A_CNT;
  float* w       = (float*)d_ws;
  float* sl1     = w + 0 * NA;
  float* con     = w + 1 * NA;
  float* con_neg = w + 2 * NA;
  float* validf  = w + 3 * NA;
  float* pL = w + 4 * NA, *pT = w + 5 * NA, *pR = w + 6 * NA, *pB = w + 7 * NA;
  float* gL = w + 8 * NA, *gT = w + 9 * NA, *gR = w + 10 * NA, *gB = w + 11 * NA;
  float* iou_s   = w + 12 * NA;
  int*   pos_num = (int*)(w + 13 * NA);
  float* sl1part = w + 13 * NA + N_B;
  float* clpart  = sl1part + (size_t)N_B * NB3;

  init_kernel<<<1, 64, 0, stream>>>(pos_num);
  prep_kernel<<<dim3(NB3, N_B), 256, 0, stream>>>(
      ploc, plabel, gloc, glabel, dboxes,
      sl1, con, con_neg, validf, pL, pT, pR, pB, gL, gT, gR, gB, pos_num);
  iou_kernel<<<dim3(NB2, N_B), 256, 0, stream>>>(pL, pT, pR, pB, gL, gT, gR, gB, iou_s);
  rank_loss_kernel<<<dim3(NB3, N_B), 256, 0, stream>>>(
      con, con_neg, sl1, validf, iou_s, glabel, pos_num, sl1part, clpart);
  final_kernel<<<1, 32, 0, stream>>>(pos_num, sl1part, clpart, out);
}